// WNO1DEncoder_87411174409096
// MI455X (gfx1250) — compile-verified
//
#include <hip/hip_runtime.h>
#include <hip/hip_bf16.h>
#include <math.h>

// ---------------------------------------------------------------------------
// Problem constants (fixed by the reference setup_inputs)
// ---------------------------------------------------------------------------
#define B_  16
#define L_  4096
#define C_  64
#define H_  128

typedef float v2f __attribute__((ext_vector_type(2)));
typedef float v8f __attribute__((ext_vector_type(8)));

__constant__ float c_lo[8] = {
  -0.010597401784997278f,  0.032883011666982945f,  0.030841381835986965f,
  -0.18703481171888114f,  -0.02798376941698385f,   0.6308807679295904f,
   0.7148465705525415f,    0.23037781330885523f };
__constant__ float c_hi[8] = {
  -0.23037781330885523f,   0.7148465705525415f,   -0.6308807679295904f,
  -0.02798376941698385f,   0.18703481171888114f,   0.030841381835986965f,
  -0.032883011666982945f, -0.010597401784997278f };

// ---------------------------------------------------------------------------
// K0: FiLM parameter precompute. grid = 4 levels * 16 batches, 256 threads.
// gamma/beta: [lev][b][h], gate: [lev][b]
// ---------------------------------------------------------------------------
__global__ __launch_bounds__(256) void k_film(
    const float* __restrict__ flow, const float* __restrict__ Wgb,
    const float* __restrict__ bgb,  const float* __restrict__ Wg,
    const float* __restrict__ bg,
    float* __restrict__ gamma_, float* __restrict__ beta_, float* __restrict__ gate_)
{
  const int lev = blockIdx.x >> 4;
  const int b   = blockIdx.x & 15;
  const int t   = threadIdx.x;
  __shared__ float fl[24];
  if (t < 24) fl[t] = flow[b * 24 + t];
  __syncthreads();
  float s = bgb[lev * 256 + t];
  const float* w = Wgb + ((size_t)lev * 256 + t) * 24;
  #pragma unroll
  for (int k = 0; k < 24; ++k) s += w[k] * fl[k];
  if (t < H_) gamma_[(lev * B_ + b) * H_ + t] = s;
  else        beta_ [(lev * B_ + b) * H_ + (t - H_)] = s;
  if (t == 0) {
    float g = bg[lev];
    const float* wg = Wg + lev * 24;
    #pragma unroll
    for (int k = 0; k < 24; ++k) g += wg[k] * fl[k];
    gate_[lev * B_ + b] = 1.0f / (1.0f + expf(-g));
  }
}

// ---------------------------------------------------------------------------
// K1: z[b,h,l] = W_in(128x64) @ (x^T * mask) + b_in, exact fp32 WMMA 16x16x4.
// One wave -> 16(M) x 64(N) tile; A fragment reused across 4 accumulators.
// ---------------------------------------------------------------------------
__global__ __launch_bounds__(256) void k_gemm_in(
    const float* __restrict__ x, const float* __restrict__ mask,
    const float* __restrict__ W_in, const float* __restrict__ b_in,
    float* __restrict__ z)
{
  const int lane  = threadIdx.x & 31;
  const int wave  = blockIdx.x * 8 + (threadIdx.x >> 5);
  const int NT    = L_ / 64;                 // 64 super-tiles of 64 columns
  const int MT    = H_ / 16;                 // 8
  const int b     = wave / (MT * NT);
  const int rem   = wave % (MT * NT);
  const int mt    = rem / NT;
  const int nt    = rem % NT;
  const int ml    = lane & 15;
  const int khalf = (lane >> 4) << 1;        // 0 or 2 (A/B K-pair select)
  const int rbase = (lane >> 4) << 3;        // 0 or 8 (C/D row-half select)

  v8f acc[4];
  #pragma unroll
  for (int s = 0; s < 4; ++s)
    #pragma unroll
    for (int r = 0; r < 8; ++r)
      acc[s][r] = b_in[mt * 16 + rbase + r];

  float mk[4]; const float* xb[4];
  #pragma unroll
  for (int s = 0; s < 4; ++s) {
    const int n = nt * 64 + s * 16 + ml;
    mk[s] = mask[b * L_ + n];
    xb[s] = x + ((size_t)b * L_ + n) * C_;
  }
  const float* Wr = W_in + (size_t)(mt * 16 + ml) * C_ + khalf;

  for (int k = 0; k < C_; k += 4) {
    v2f a; a.x = Wr[k]; a.y = Wr[k + 1];
    #pragma unroll
    for (int s = 0; s < 4; ++s) {
      v2f bf; bf.x = xb[s][k + khalf] * mk[s]; bf.y = xb[s][k + khalf + 1] * mk[s];
      acc[s] = __builtin_amdgcn_wmma_f32_16x16x4_f32(
          false, a, false, bf, (short)0, acc[s], false, false);
    }
  }

  #pragma unroll
  for (int s = 0; s < 4; ++s) {
    const int n = nt * 64 + s * 16 + ml;
    float* zp = z + ((size_t)b * H_ + mt * 16 + rbase) * L_ + n;
    #pragma unroll
    for (int r = 0; r < 8; ++r) zp[(size_t)r * L_] = acc[s][r];
  }
}

// ---------------------------------------------------------------------------
// K2: per-(b,h) row: 3-level circular DWT, FiLM, soft-threshold, inverse DWT.
// Entire row pipeline lives in LDS (~42 KB of the 320 KB/WGP).
// ---------------------------------------------------------------------------
__device__ __forceinline__ void analysis(const float* __restrict__ src, int n,
                                         float* __restrict__ lo, float* __restrict__ hi,
                                         int tid)
{
  const int half = n >> 1;
  for (int i = tid; i < half; i += 256) {
    float a = 0.f, d = 0.f;
    const int j = i << 1;
    #pragma unroll
    for (int k = 0; k < 8; ++k) {
      const float v = src[(j + k) & (n - 1)];
      a += c_lo[k] * v;
      d += c_hi[k] * v;
    }
    lo[i] = a; hi[i] = d;
  }
}

__device__ __forceinline__ void synth_pair(const float* __restrict__ a,
                                           const float* __restrict__ d, int n,
                                           float* __restrict__ out, int tid)
{
  const int N = n << 1;
  for (int j = tid; j < N; j += 256) {
    float s = 0.f;
    #pragma unroll
    for (int k = (j & 1); k < 8; k += 2) {   // only (j-k) even hits nonzero taps
      const int idx = ((j - k) & (N - 1)) >> 1;
      s += c_lo[k] * a[idx] + c_hi[k] * d[idx];
    }
    out[j] = s;
  }
}

__device__ __forceinline__ float film_thresh(float* __restrict__ arr, int len,
                                             float s1, float s2, float lamscale,
                                             float* __restrict__ red, int tid)
{
  for (int i = tid; i < len; i += 256) arr[i] = arr[i] * s1 + s2;
  __syncthreads();
  float acc = 0.f;
  for (int i = tid; i < len; i += 256) acc += fabsf(arr[i]);
  red[tid] = acc; __syncthreads();
  for (int o = 128; o > 0; o >>= 1) {
    if (tid < o) red[tid] += red[tid + o];
    __syncthreads();
  }
  const float lam = lamscale * (red[0] / (float)len);
  for (int i = tid; i < len; i += 256) {
    const float v  = arr[i];
    const float av = fabsf(v) - lam;
    arr[i] = (av > 0.f) ? copysignf(av, v) : 0.f;
  }
  __syncthreads();
  return lam;
}

__global__ __launch_bounds__(256) void k_wavelet(
    const float* __restrict__ z,
    const float* __restrict__ gamma_, const float* __restrict__ beta_,
    const float* __restrict__ gate_,  const float* __restrict__ lam_raw,
    float* __restrict__ zrec, float* __restrict__ lamws)
{
  __shared__ float cur[4096];
  __shared__ float tmp[2048];
  __shared__ float d1[2048];
  __shared__ float d2[1024];
  __shared__ float d3[512];
  __shared__ float ap[512];
  __shared__ float red[256];

  const int tid = threadIdx.x;
  const int b   = blockIdx.x >> 7;
  const int h   = blockIdx.x & 127;

  const float* zr = z + ((size_t)b * H_ + h) * L_;
  for (int i = tid; i < L_; i += 256) cur[i] = zr[i];
  __syncthreads();

  analysis(cur, 4096, tmp, d1, tid); __syncthreads();   // tmp = a1(2048)
  analysis(tmp, 2048, cur, d2, tid); __syncthreads();   // cur[0:1024) = a2
  analysis(cur, 1024, ap,  d3, tid); __syncthreads();   // ap = a3(512)

  float lam[4];
  float* arrs[4] = { ap, d1, d2, d3 };                  // FiLM/lam index == slot
  const int lens[4] = { 512, 2048, 1024, 512 };
  #pragma unroll
  for (int s = 0; s < 4; ++s) {
    const float g   = gate_[s * B_ + b];
    const float s1  = 1.0f + g * gamma_[(s * B_ + b) * H_ + h];
    const float s2  = g * beta_[(s * B_ + b) * H_ + h];
    const float lsc = log1pf(expf(lam_raw[s]));         // softplus
    lam[s] = film_thresh(arrs[s], lens[s], s1, s2, lsc, red, tid);
  }
  if (tid == 0) {
    #pragma unroll
    for (int s = 0; s < 4; ++s) lamws[(size_t)b * 512 + s * 128 + h] = lam[s];
  }

  synth_pair(ap,  d3, 512,  cur, tid); __syncthreads(); // cur[0:1024)
  synth_pair(cur, d2, 1024, tmp, tid); __syncthreads(); // tmp[0:2048)
  synth_pair(tmp, d1, 2048, cur, tid); __syncthreads(); // cur[0:4096)

  float* zo = zrec + ((size_t)b * H_ + h) * L_;
  for (int i = tid; i < L_; i += 256) zo[i] = cur[i];
}

// ---------------------------------------------------------------------------
// K3: mixed = W_mix(128x128) @ (z + z_rec), bias, exact GELU, mask, transposed
// store to h_seq[b,l,h]. fp32 WMMA, K=128.
// ---------------------------------------------------------------------------
__global__ __launch_bounds__(256) void k_gemm_mix(
    const float* __restrict__ z, const float* __restrict__ zrec,
    const float* __restrict__ mask,
    const float* __restrict__ W_mix, const float* __restrict__ b_mix,
    float* __restrict__ hseq)
{
  const int lane  = threadIdx.x & 31;
  const int wave  = blockIdx.x * 8 + (threadIdx.x >> 5);
  const int NT    = L_ / 64;
  const int MT    = H_ / 16;
  const int b     = wave / (MT * NT);
  const int rem   = wave % (MT * NT);
  const int mt    = rem / NT;
  const int nt    = rem % NT;
  const int ml    = lane & 15;
  const int khalf = (lane >> 4) << 1;
  const int rbase = (lane >> 4) << 3;

  v8f acc[4];
  #pragma unroll
  for (int s = 0; s < 4; ++s)
    #pragma unroll
    for (int r = 0; r < 8; ++r)
      acc[s][r] = b_mix[mt * 16 + rbase + r];

  int ncol[4];
  #pragma unroll
  for (int s = 0; s < 4; ++s) ncol[s] = nt * 64 + s * 16 + ml;

  const float* Ar = W_mix + (size_t)(mt * 16 + ml) * H_ + khalf;
  const float* zb = z    + (size_t)b * H_ * L_;
  const float* rb = zrec + (size_t)b * H_ * L_;

  for (int k = 0; k < H_; k += 4) {
    v2f a; a.x = Ar[k]; a.y = Ar[k + 1];
    const int kk = k + khalf;
    const float* z0 = zb + (size_t)kk * L_;
    const float* z1 = zb + (size_t)(kk + 1) * L_;
    const float* r0 = rb + (size_t)kk * L_;
    const float* r1 = rb + (size_t)(kk + 1) * L_;
    #pragma unroll
    for (int s = 0; s < 4; ++s) {
      v2f bf; bf.x = z0[ncol[s]] + r0[ncol[s]]; bf.y = z1[ncol[s]] + r1[ncol[s]];
      acc[s] = __builtin_amdgcn_wmma_f32_16x16x4_f32(
          false, a, false, bf, (short)0, acc[s], false, false);
    }
  }

  #pragma unroll
  for (int s = 0; s < 4; ++s) {
    const int n = ncol[s];
    const float mk = mask[b * L_ + n];
    float* op = hseq + ((size_t)b * L_ + n) * H_ + mt * 16 + rbase;
    #pragma unroll
    for (int r = 0; r < 8; ++r) {
      float v = acc[s][r];
      v = 0.5f * v * (1.0f + erff(v * 0.7071067811865475f));   // exact GELU
      op[r] = v * mk;
    }
  }
}

// ---------------------------------------------------------------------------
// K4: masked mean pool over L per (b,h).  K5: lambda mean/max per b.
// ---------------------------------------------------------------------------
__global__ __launch_bounds__(256) void k_pool(
    const float* __restrict__ hseq, const float* __restrict__ mask,
    float* __restrict__ hpool)
{
  __shared__ float r1[256]; __shared__ float r2[256];
  const int tid = threadIdx.x;
  const int b = blockIdx.x >> 7;
  const int h = blockIdx.x & 127;
  const float* hp = hseq + (size_t)b * L_ * H_ + h;
  const float* mr = mask + (size_t)b * L_;
  float a = 0.f, m = 0.f;
  for (int l = tid; l < L_; l += 256) { a += hp[(size_t)l * H_] * mr[l]; m += mr[l]; }
  r1[tid] = a; r2[tid] = m; __syncthreads();
  for (int o = 128; o > 0; o >>= 1) {
    if (tid < o) { r1[tid] += r1[tid + o]; r2[tid] += r2[tid + o]; }
    __syncthreads();
  }
  if (tid == 0) {
    const float den = r2[0] < 1.0f ? 1.0f : r2[0];
    hpool[b * H_ + h] = r1[0] / den;
  }
}

__global__ __launch_bounds__(256) void k_stats(
    const float* __restrict__ lamws, float* __restrict__ omean, float* __restrict__ omax)
{
  __shared__ float rs[256]; __shared__ float rm[256];
  const int tid = threadIdx.x;
  const int b = blockIdx.x;
  float s = 0.f, mx = -3.4e38f;
  for (int i = tid; i < 512; i += 256) {
    const float v = lamws[(size_t)b * 512 + i];
    s += v; mx = fmaxf(mx, v);
  }
  rs[tid] = s; rm[tid] = mx; __syncthreads();
  for (int o = 128; o > 0; o >>= 1) {
    if (tid < o) { rs[tid] += rs[tid + o]; rm[tid] = fmaxf(rm[tid], rm[tid + o]); }
    __syncthreads();
  }
  if (tid == 0) { omean[b] = rs[0] / 512.0f; omax[b] = rm[0]; }
}

// ---------------------------------------------------------------------------
// Host launcher
// ---------------------------------------------------------------------------
extern "C" void kernel_launch(void* const* d_in, const int* in_sizes, int n_in,
                              void* d_out, int out_size, void* d_ws, size_t ws_size,
                              hipStream_t stream)
{
  (void)in_sizes; (void)n_in; (void)out_size; (void)ws_size;

  const float* x       = (const float*)d_in[0];
  const float* mask    = (const float*)d_in[1];
  const float* flow    = (const float*)d_in[2];
  const float* W_in    = (const float*)d_in[3];
  const float* b_in    = (const float*)d_in[4];
  const float* W_mix   = (const float*)d_in[5];
  const float* b_mix   = (const float*)d_in[6];
  const float* lam_raw = (const float*)d_in[7];
  const float* Wgb     = (const float*)d_in[8];
  const float* bgb     = (const float*)d_in[9];
  const float* Wg      = (const float*)d_in[10];
  const float* bg      = (const float*)d_in[11];

  float* out   = (float*)d_out;
  float* hseq  = out;                                  // [B,L,H]
  float* hpool = out + (size_t)B_ * L_ * H_;           // [B,H]
  float* omean = hpool + (size_t)B_ * H_;              // [B]
  float* omax  = omean + B_;                           // [B]

  float* ws    = (float*)d_ws;
  float* z     = ws;                                   // [B,H,L]
  float* zrec  = z + (size_t)B_ * H_ * L_;             // [B,H,L]
  float* gamma = zrec + (size_t)B_ * H_ * L_;          // [4,B,H]
  float* beta  = gamma + (size_t)4 * B_ * H_;          // [4,B,H]
  float* gate  = beta + (size_t)4 * B_ * H_;           // [4,B]
  float* lamws = gate + (size_t)4 * B_;                // [B,4,H]

  k_film    <<<4 * B_,   256, 0, stream>>>(flow, Wgb, bgb, Wg, bg, gamma, beta, gate);
  k_gemm_in <<<1024,     256, 0, stream>>>(x, mask, W_in, b_in, z);
  k_wavelet <<<B_ * H_,  256, 0, stream>>>(z, gamma, beta, gate, lam_raw, zrec, lamws);
  k_gemm_mix<<<1024,     256, 0, stream>>>(z, zrec, mask, W_mix, b_mix, hseq);
  k_pool    <<<B_ * H_,  256, 0, stream>>>(hseq, mask, hpool);
  k_stats   <<<B_,       256, 0, stream>>>(lamws, omean, omax);
}